// MinimalDifferentiableTensorSketch_21818433864270
// MI455X (gfx1250) — compile-verified
//
#include <hip/hip_runtime.h>

typedef __attribute__((ext_vector_type(2))) float v2f;
typedef __attribute__((ext_vector_type(8))) float v8f;

#define NB   1024   // histogram blocks (partials: NB*4 uints = 16 KB of d_ws)
#define TPB  256    // 8 waves per block on wave32

__device__ __forceinline__ void accum_tok(unsigned long long& acc, int tok) {
    unsigned v = (unsigned)tok;            // c==4 or negative -> invalid, masked
    if (v < 4u) acc += 1ull << (v << 4);   // four 16-bit counters packed in u64
}

__device__ __forceinline__ void accum_vec(unsigned long long& acc, int4 p) {
    accum_tok(acc, p.x); accum_tok(acc, p.y);
    accum_tok(acc, p.z); accum_tok(acc, p.w);
}

// Stage 1: streaming histogram of symbols {0,1,2,3} over the token stream.
__global__ void sketch_hist_kernel(const int* __restrict__ seq, long long n,
                                   unsigned* __restrict__ partials) {
    __shared__ unsigned sh[4];
    if (threadIdx.x < 4) sh[threadIdx.x] = 0u;
    __syncthreads();

    const long long gid  = (long long)blockIdx.x * blockDim.x + threadIdx.x;
    const long long T    = (long long)gridDim.x * blockDim.x;
    const long long nvec = n >> 2;
    const int4* __restrict__ vp = (const int4*)seq;

    unsigned long long acc = 0ull;
    long long i = gid;
    // 4 independent B128 loads per iteration -> deep MLP for HBM latency
    for (; i + 3 * T < nvec; i += 4 * T) {
        int4 p0 = vp[i];
        int4 p1 = vp[i + T];
        int4 p2 = vp[i + 2 * T];
        int4 p3 = vp[i + 3 * T];
        accum_vec(acc, p0); accum_vec(acc, p1);
        accum_vec(acc, p2); accum_vec(acc, p3);
    }
    for (; i < nvec; i += T) accum_vec(acc, vp[i]);
    // scalar tail (n not multiple of 4)
    long long rem = n & 3;
    if (gid < rem) accum_tok(acc, seq[(nvec << 2) + gid]);

    unsigned c0 = (unsigned)( acc        & 0xFFFFull);
    unsigned c1 = (unsigned)((acc >> 16) & 0xFFFFull);
    unsigned c2 = (unsigned)((acc >> 32) & 0xFFFFull);
    unsigned c3 = (unsigned)((acc >> 48) & 0xFFFFull);

    // wave32 reduction
    for (int off = 16; off > 0; off >>= 1) {
        c0 += __shfl_down(c0, off, 32);
        c1 += __shfl_down(c1, off, 32);
        c2 += __shfl_down(c2, off, 32);
        c3 += __shfl_down(c3, off, 32);
    }
    if ((threadIdx.x & 31) == 0) {
        atomicAdd(&sh[0], c0); atomicAdd(&sh[1], c1);
        atomicAdd(&sh[2], c2); atomicAdd(&sh[3], c3);
    }
    __syncthreads();
    if (threadIdx.x < 4)
        partials[(long long)blockIdx.x * 4 + threadIdx.x] = sh[threadIdx.x];
}

// Stage 2 (one wave32): reduce partial counts, then
// out[0:16] = row 0 of (A[16x4] x B[4x16]) via V_WMMA_F32_16X16X4_F32,
// A row0[c] = count[c]*tanh(sign[c]), B = hash_embedding.
__global__ void sketch_finalize_kernel(const unsigned* __restrict__ partials,
                                       const float* __restrict__ emb,
                                       const float* __restrict__ sgn,
                                       float* __restrict__ out, int nb) {
    const int lane = threadIdx.x;  // 32 threads, EXEC all ones

    unsigned a0 = 0, a1 = 0, a2 = 0, a3 = 0;
    const uint4* __restrict__ pv = (const uint4*)partials;
    for (int b = lane; b < nb; b += 32) {
        uint4 p = pv[b];
        a0 += p.x; a1 += p.y; a2 += p.z; a3 += p.w;
    }
    // xor-reduce: every lane ends with all four totals
    for (int off = 16; off > 0; off >>= 1) {
        a0 += __shfl_xor(a0, off, 32);
        a1 += __shfl_xor(a1, off, 32);
        a2 += __shfl_xor(a2, off, 32);
        a3 += __shfl_xor(a3, off, 32);
    }

    const float w0 = (float)a0 * tanhf(sgn[0]);
    const float w1 = (float)a1 * tanhf(sgn[1]);
    const float w2 = (float)a2 * tanhf(sgn[2]);
    const float w3 = (float)a3 * tanhf(sgn[3]);

    // A-matrix 16x4 f32 layout: lanes 0-15 -> M=lane, {v0,v1}={K0,K1};
    //                           lanes 16-31 -> M=lane-16, {v0,v1}={K2,K3}.
    // B-matrix 4x16 f32 layout: lanes 0-15 -> N=lane, {v0,v1}={K0,K1};
    //                           lanes 16-31 -> N=lane-16, {v0,v1}={K2,K3}.
    const int  m     = lane & 15;
    const bool lowK  = (lane < 16);
    const float wA0  = lowK ? w0 : w2;
    const float wA1  = lowK ? w1 : w3;
    const int   k0   = lowK ? 0 : 2;

    v2f A;
    A.x = (m == 0) ? wA0 : 0.0f;
    A.y = (m == 0) ? wA1 : 0.0f;
    v2f B;
    B.x = emb[k0 * 16 + m];
    B.y = emb[(k0 + 1) * 16 + m];

    v8f C = {};
#if defined(__has_builtin) && __has_builtin(__builtin_amdgcn_wmma_f32_16x16x4_f32)
    // 8 args: (neg_a, A, neg_b, B, c_mod, C, reuse_a, reuse_b)
    C = __builtin_amdgcn_wmma_f32_16x16x4_f32(
            false, A, false, B, (short)0, C, false, false);
    // D row M=0 lives in VGPR0 on lanes 0..15 (N = lane)
    if (lane < 16) out[lane] = C[0];
#else
    // fallback (should not trigger on gfx1250): scalar combine
    if (lane < 16)
        out[lane] = w0 * emb[0 * 16 + lane] + w1 * emb[1 * 16 + lane] +
                    w2 * emb[2 * 16 + lane] + w3 * emb[3 * 16 + lane];
#endif
}

extern "C" void kernel_launch(void* const* d_in, const int* in_sizes, int n_in,
                              void* d_out, int out_size, void* d_ws, size_t ws_size,
                              hipStream_t stream) {
    const int*   seq = (const int*)d_in[0];    // sequence, int32
    const float* emb = (const float*)d_in[1];  // hash_embedding [4,16] f32
    const float* sgn = (const float*)d_in[2];  // sign_weight [4] f32
    float*       out = (float*)d_out;          // [16] f32
    const long long n = (long long)in_sizes[0];
    unsigned* partials = (unsigned*)d_ws;      // NB*4 uints = 16 KB

    sketch_hist_kernel<<<NB, TPB, 0, stream>>>(seq, n, partials);
    sketch_finalize_kernel<<<1, 32, 0, stream>>>(partials, emb, sgn, out, NB);
}